// WaveletBlock_75222057222396
// MI455X (gfx1250) — compile-verified
//
#include <hip/hip_runtime.h>

// 2D Haar wavelet (2x2 butterfly) on (B=16, C=64, H=512, W=512) fp32.
// Pure streaming kernel: ~1 GiB in, ~1 GiB out, AI ~0.4 flop/byte ->
// HBM-bound (floor ~92 us @ 23.3 TB/s). Strategy: b128 non-temporal
// loads/stores (gfx1250 TH=NT cache policy), 4 output columns per thread.

#define INV_SQRT2_SQ 0.5f

// Native clang vector type (required by __builtin_nontemporal_*; HIP's
// float4 is a struct and is rejected).
typedef float v4f __attribute__((ext_vector_type(4)));

__global__ __launch_bounds__(256) void haar2x2_nt_kernel(
    const float* __restrict__ x,   // (BC, H, W) flattened
    float*       __restrict__ out, // 4 planes of (BC, Hout, Wout), concatenated
    int quads,                     // total float4-groups per plane
    long long planeTotal)          // elements per output plane
{
    constexpr int W    = 512;
    constexpr int H    = 512;
    constexpr int Hout = H / 2;     // 256
    constexpr int Wout = W / 2;     // 256
    constexpr int WoutQ = Wout / 4; // 64 vec4 groups per output row

    int q = blockIdx.x * blockDim.x + threadIdx.x;
    if (q >= quads) return;

    int jq = q & (WoutQ - 1);      // vec4 group within output row
    int r  = q >> 6;               // q / WoutQ  -> global output row id
    int i  = r & (Hout - 1);       // output row within image
    int bc = r >> 8;               // r / Hout   -> (batch*channel) index

    // Input: rows 2i and 2i+1, columns [8*jq, 8*jq+8)
    const size_t inBase = (size_t)bc * (size_t)(H * W)
                        + (size_t)(2 * i) * (size_t)W
                        + (size_t)(8 * jq);
    const v4f* pTop = reinterpret_cast<const v4f*>(x + inBase);
    const v4f* pBot = reinterpret_cast<const v4f*>(x + inBase + W);

    v4f t0 = __builtin_nontemporal_load(pTop);       // cols 8jq..8jq+3 (even row)
    v4f t1 = __builtin_nontemporal_load(pTop + 1);   // cols 8jq+4..8jq+7
    v4f b0 = __builtin_nontemporal_load(pBot);       // odd row
    v4f b1 = __builtin_nontemporal_load(pBot + 1);

    v4f vA, vH, vV, vD;

    // pair k: a = even-row even-col, b = even-row odd-col,
    //         c = odd-row  even-col, d = odd-row  odd-col
    #define BUTTERFLY(K, A, B, C, D)                               \
        {                                                          \
            float sab = (A) + (B), scd = (C) + (D);                \
            float dab = (A) - (B), dcd = (C) - (D);                \
            vA[K] = (sab + scd) * INV_SQRT2_SQ;                    \
            vH[K] = (sab - scd) * INV_SQRT2_SQ;                    \
            vV[K] = (dab + dcd) * INV_SQRT2_SQ;                    \
            vD[K] = (dab - dcd) * INV_SQRT2_SQ;                    \
        }

    BUTTERFLY(0, t0[0], t0[1], b0[0], b0[1])
    BUTTERFLY(1, t0[2], t0[3], b0[2], b0[3])
    BUTTERFLY(2, t1[0], t1[1], b1[0], b1[1])
    BUTTERFLY(3, t1[2], t1[3], b1[2], b1[3])
    #undef BUTTERFLY

    const size_t outIdx = (size_t)bc * (size_t)(Hout * Wout)
                        + (size_t)i * (size_t)Wout
                        + (size_t)(4 * jq);

    float* oA = out + outIdx;
    float* oH = oA + (size_t)planeTotal;
    float* oV = oH + (size_t)planeTotal;
    float* oD = oV + (size_t)planeTotal;

    __builtin_nontemporal_store(vA, reinterpret_cast<v4f*>(oA));
    __builtin_nontemporal_store(vH, reinterpret_cast<v4f*>(oH));
    __builtin_nontemporal_store(vV, reinterpret_cast<v4f*>(oV));
    __builtin_nontemporal_store(vD, reinterpret_cast<v4f*>(oD));
}

extern "C" void kernel_launch(void* const* d_in, const int* in_sizes, int n_in,
                              void* d_out, int out_size, void* d_ws, size_t ws_size,
                              hipStream_t stream) {
    (void)n_in; (void)d_ws; (void)ws_size; (void)out_size;

    const float* x = (const float*)d_in[0];
    float* out = (float*)d_out;

    // n = B*C*H*W; each 2x2 input block -> 1 element in each of 4 planes.
    const long long n = (long long)in_sizes[0];
    const long long planeTotal = n / 4;       // elements per output plane
    const int quads = (int)(planeTotal / 4);  // vec4 groups per plane

    const int threads = 256;
    const int blocks = (quads + threads - 1) / threads;

    haar2x2_nt_kernel<<<blocks, threads, 0, stream>>>(x, out, quads, planeTotal);
}